// EdgeConv_21019569947180
// MI455X (gfx1250) — compile-verified
//
#include <hip/hip_runtime.h>

typedef __attribute__((ext_vector_type(2))) float v2f;
typedef __attribute__((ext_vector_type(8))) float v8f;

#define N_NODES 100000
#define N_EDGES 1600000
#define DIN 48
#define DOUT 48
#define DCAT 96          // [A | C] concatenated per node
#define NTILES (N_NODES / 16)   // 6250, exact

// ---------------- prep: Wcat = [theta ; phi - theta], bias_c = phi_b + theta_b ----
__global__ void ec_prep(const float* __restrict__ theta_w, const float* __restrict__ theta_b,
                        const float* __restrict__ phi_w,   const float* __restrict__ phi_b,
                        float* __restrict__ Wcat, float* __restrict__ bias_c) {
  int i = blockIdx.x * blockDim.x + threadIdx.x;
  if (i < DOUT * DIN) {
    Wcat[i] = theta_w[i];                          // rows 0..47  -> A weights
    Wcat[DOUT * DIN + i] = phi_w[i] - theta_w[i];  // rows 48..95 -> C weights
  }
  if (i < DOUT) bias_c[i] = phi_b[i] + theta_b[i];
}

// ---------------- init scatter-max keys to sentinel 0 (== encoded -NaN) ----------
__global__ void ec_init(unsigned* __restrict__ Mkey) {
  int i = blockIdx.x * blockDim.x + threadIdx.x;
  if (i < N_NODES * DOUT) Mkey[i] = 0u;
}

// ---------------- per-node GEMM: AC = h @ Wcat^T (+bias on C half) via f32 WMMA --
// One wave per 16-node tile. A (16x4 f32) layout: lane%16 = M, lane/16 = K-half,
// VGPR0/1 = consecutive K -> one float2 load per step. B (4x16) mirrors it.
__global__ void ec_gemm(const float* __restrict__ h,
                        const float* __restrict__ Wcat,
                        const float* __restrict__ bias_c,
                        float* __restrict__ AC) {
  const int lane  = threadIdx.x & 31;
  const int wave  = threadIdx.x >> 5;
  const int tile  = blockIdx.x * (blockDim.x >> 5) + wave;
  if (tile >= NTILES) return;                 // wave-uniform: EXEC stays all-ones
  const int row   = lane & 15;                // M for A, N for B/C/D
  const int khalf = lane >> 4;                // 0: K={0,1}, 1: K={2,3} per step

  const float* hrow = h + (tile * 16 + row) * DIN + khalf * 2;
  v2f a[12];
#pragma unroll
  for (int kk = 0; kk < 12; ++kk)
    a[kk] = *(const v2f*)(hrow + kk * 4);

#pragma unroll
  for (int nt = 0; nt < 6; ++nt) {
    const float* wrow = Wcat + (nt * 16 + row) * DIN + khalf * 2;
    v8f c = {};
#pragma unroll
    for (int kk = 0; kk < 12; ++kk) {
      v2f b = *(const v2f*)(wrow + kk * 4);
      c = __builtin_amdgcn_wmma_f32_16x16x4_f32(false, a[kk], false, b,
                                                (short)0, c, false, false);
    }
    const float bias = (nt >= 3) ? bias_c[nt * 16 + row - DOUT] : 0.0f;
#pragma unroll
    for (int v = 0; v < 8; ++v) {
      const int m = v + 8 * khalf;            // D row: VGPR v, lanes16-31 -> +8
      AC[(tile * 16 + m) * DCAT + nt * 16 + row] = c[v] + bias;
    }
  }
}

// ---------------- scatter-max over edges: Mkey[dst] = max(Mkey[dst], key(C[src])) -
// Monotonic uint encoding of f32 -> native global_atomic_max_u32.
__device__ __forceinline__ unsigned ec_fkey(float f) {
  unsigned u = __float_as_uint(f);
  return (u & 0x80000000u) ? ~u : (u | 0x80000000u);
}

__global__ void ec_edges(const int* __restrict__ src, const int* __restrict__ dst,
                         const float* __restrict__ AC, unsigned* __restrict__ Mkey) {
  int t = blockIdx.x * blockDim.x + threadIdx.x;
  int e = t >> 4;                 // 16 lanes per edge
  int g = t & 15;                 // each lane covers 3 features (48 total)
  if (e >= N_EDGES) return;
  int s = src[e];
  int d = dst[e];
  const float* crow = AC + (size_t)s * DCAT + DOUT + g * 3;  // C half of source row
  unsigned* mrow = Mkey + (size_t)d * DOUT + g * 3;
#pragma unroll
  for (int j = 0; j < 3; ++j)
    atomicMax(&mrow[j], ec_fkey(crow[j]));
}

// ---------------- finalize: out = (no in-edge) ? 0 : A[d] + decode(max-key) -------
__global__ void ec_final(const float* __restrict__ AC, const unsigned* __restrict__ Mkey,
                         float* __restrict__ out) {
  int i = blockIdx.x * blockDim.x + threadIdx.x;
  if (i >= N_NODES * DOUT) return;
  int n = i / DOUT;
  int f = i - n * DOUT;
  unsigned k = Mkey[i];
  float m = (k & 0x80000000u) ? __uint_as_float(k & 0x7FFFFFFFu)
                              : __uint_as_float(~k);
  out[i] = (k == 0u) ? 0.0f : (AC[(size_t)n * DCAT + f] + m);
}

extern "C" void kernel_launch(void* const* d_in, const int* in_sizes, int n_in,
                              void* d_out, int out_size, void* d_ws, size_t ws_size,
                              hipStream_t stream) {
  const float* h       = (const float*)d_in[0];
  const float* theta_w = (const float*)d_in[1];
  const float* theta_b = (const float*)d_in[2];
  const float* phi_w   = (const float*)d_in[3];
  const float* phi_b   = (const float*)d_in[4];
  const int*   src     = (const int*)d_in[5];
  const int*   dst     = (const int*)d_in[6];
  float* out = (float*)d_out;

  // workspace layout
  char* ws = (char*)d_ws;
  float*    Wcat   = (float*)ws;                                   // 96*48*4 = 18432 B
  float*    bias_c = (float*)(ws + 18432);                         // 192 B
  float*    AC     = (float*)(ws + 18688);                         // N*96*4 = 38.4 MB
  unsigned* Mkey   = (unsigned*)(ws + 18688 + (size_t)N_NODES * DCAT * 4); // N*48*4 = 19.2 MB

  // 1) fold weights/biases
  ec_prep<<<(DOUT * DIN + 255) / 256, 256, 0, stream>>>(theta_w, theta_b, phi_w, phi_b,
                                                        Wcat, bias_c);
  // 2) init scatter-max keys
  ec_init<<<(N_NODES * DOUT + 255) / 256, 256, 0, stream>>>(Mkey);
  // 3) per-node GEMM via f32 WMMA: 8 waves/block, 1 tile/wave
  ec_gemm<<<(NTILES + 7) / 8, 256, 0, stream>>>(h, Wcat, bias_c, AC);
  // 4) edge scatter-max (16 threads per edge)
  ec_edges<<<(N_EDGES * 16) / 256, 256, 0, stream>>>(src, dst, AC, Mkey);
  // 5) combine
  ec_final<<<(N_NODES * DOUT + 255) / 256, 256, 0, stream>>>(AC, Mkey, out);
}